// GraphConvolutionSparse_84980222918784
// MI455X (gfx1250) — compile-verified
//
#include <hip/hip_runtime.h>

typedef __attribute__((ext_vector_type(2))) float v2f;
typedef __attribute__((ext_vector_type(4))) float v4f;
typedef __attribute__((ext_vector_type(8))) float v8f;

#define F_IN  512
#define F_OUT 256
#define EPB   16   // edges per SpMM block

// ---------------------------------------------------------------------------
// GEMM: h[M,256] = x[M,512] @ w[512,256], fp32 via V_WMMA_F32_16X16X4_F32.
// Block = 128 threads = 4 waves, block tile = 64 rows x 256 cols.
// Wave w owns cols [w*64, w*64+64) and ALL 4 row-subtiles:
//   16 accumulators (4 M-tiles x 4 N-tiles) -> per k-step,
//   4 A b64 loads + 8 B b32 loads feed 16 WMMAs (4x the FLOP/byte of the
//   16-row version; W re-read traffic from L2 drops 4x to ~0.8 GB).
//
// A-frag (16x4 f32): lane l<16 holds A[l][k+0..1], lane l>=16 A[l-16][k+2..3].
// B-frag (4x16 f32): vgpr j, lane-half hh: B[k + hh*2 + j][n0 + lane%16].
// C/D: vgpr i: lanes 0-15 -> (M=i, N=lane), lanes 16-31 -> (M=i+8, N=lane-16).
// ---------------------------------------------------------------------------
__global__ __launch_bounds__(128) void gcn_gemm_wmma(
    const float* __restrict__ x, const float* __restrict__ w,
    float* __restrict__ h, int M) {
  const int wave  = threadIdx.x >> 5;
  const int lane  = threadIdx.x & 31;
  const int lrow  = lane & 15;
  const int khalf = (lane >> 4) << 1;       // 0 or 2
  const int m0    = blockIdx.x << 6;        // 64 rows per block
  const int n0    = wave << 6;              // 64 cols per wave

  // Clamped A row pointers (keep EXEC all-1s inside the WMMA loop).
  const float* __restrict__ ap[4];
#pragma unroll
  for (int mt = 0; mt < 4; ++mt) {
    int r = m0 + mt * 16 + lrow;
    if (r >= M) r = M - 1;
    ap[mt] = x + (size_t)r * F_IN + khalf;
  }
  const float* __restrict__ bp = w + (size_t)khalf * F_OUT + n0 + lrow;

  v8f acc[4][4];
#pragma unroll
  for (int mt = 0; mt < 4; ++mt)
#pragma unroll
    for (int nt = 0; nt < 4; ++nt) acc[mt][nt] = (v8f){};

#pragma unroll 2
  for (int k = 0; k < F_IN; k += 4) {
    v2f a[4];
#pragma unroll
    for (int mt = 0; mt < 4; ++mt) a[mt] = *(const v2f*)(ap[mt] + k);

    const float* __restrict__ b = bp + (size_t)k * F_OUT;
    v2f bf[4];
#pragma unroll
    for (int nt = 0; nt < 4; ++nt) {
      bf[nt].x = b[nt * 16];
      bf[nt].y = b[F_OUT + nt * 16];
    }

#pragma unroll
    for (int mt = 0; mt < 4; ++mt)
#pragma unroll
      for (int nt = 0; nt < 4; ++nt)
        acc[mt][nt] = __builtin_amdgcn_wmma_f32_16x16x4_f32(
            false, a[mt], false, bf[nt], (short)0, acc[mt][nt], false, false);
  }

  const int crow = (lane >> 4) << 3;        // 0 or 8
#pragma unroll
  for (int mt = 0; mt < 4; ++mt) {
    float* __restrict__ hp =
        h + (size_t)(m0 + mt * 16 + crow) * F_OUT + n0 + lrow;
#pragma unroll
    for (int i = 0; i < 8; ++i) {
      if (m0 + mt * 16 + crow + i < M) {
        hp[(size_t)i * F_OUT +  0] = acc[mt][0][i];
        hp[(size_t)i * F_OUT + 16] = acc[mt][1][i];
        hp[(size_t)i * F_OUT + 32] = acc[mt][2][i];
        hp[(size_t)i * F_OUT + 48] = acc[mt][3][i];
      }
    }
  }
}

// ---------------------------------------------------------------------------
// SpMM: out[rows[e], :] += vals[e] * h[cols[e], :]. threadIdx.x == feature.
// h (102 MB) is L2-resident on MI455X (192 MB L2); gathers and atomics are
// fully coalesced 1KB lines. Edge metadata is uniform per iteration.
// ---------------------------------------------------------------------------
__global__ __launch_bounds__(256) void gcn_spmm_atomic(
    const int* __restrict__ rows, const int* __restrict__ cols,
    const float* __restrict__ vals, const float* __restrict__ h,
    float* __restrict__ out, int E) {
  const int f = threadIdx.x;                // 0..255
  int e = blockIdx.x * EPB;
  int eend = e + EPB; if (eend > E) eend = E;
  for (; e < eend; ++e) {
    const int   r = rows[e];
    const int   c = cols[e];
    const float v = vals[e];
    const float m = v * h[(size_t)c * F_OUT + f];
    __hip_atomic_fetch_add(&out[(size_t)r * F_OUT + f], m,
                           __ATOMIC_RELAXED, __HIP_MEMORY_SCOPE_AGENT);
  }
}

// ---------------------------------------------------------------------------
__global__ void gcn_zero(v4f* __restrict__ p, long n4) {
  long i = (long)blockIdx.x * blockDim.x + threadIdx.x;
  long stride = (long)gridDim.x * blockDim.x;
  v4f z = {};
  for (; i < n4; i += stride) p[i] = z;
}

__global__ void gcn_relu(v4f* __restrict__ p, long n4) {
  long i = (long)blockIdx.x * blockDim.x + threadIdx.x;
  long stride = (long)gridDim.x * blockDim.x;
  for (; i < n4; i += stride) {
    v4f t = p[i];
    t.x = t.x > 0.f ? t.x : 0.f;
    t.y = t.y > 0.f ? t.y : 0.f;
    t.z = t.z > 0.f ? t.z : 0.f;
    t.w = t.w > 0.f ? t.w : 0.f;
    p[i] = t;
  }
}

// ---------------------------------------------------------------------------
extern "C" void kernel_launch(void* const* d_in, const int* in_sizes, int n_in,
                              void* d_out, int out_size, void* d_ws, size_t ws_size,
                              hipStream_t stream) {
  const float* x    = (const float*)d_in[0];
  const int*   rows = (const int*)  d_in[1];
  const int*   cols = (const int*)  d_in[2];
  const float* vals = (const float*)d_in[3];
  const float* w    = (const float*)d_in[4];
  float*       out  = (float*)d_out;
  float*       h    = (float*)d_ws;            // M*F_OUT fp32 scratch

  const int M = in_sizes[0] / F_IN;            // 100000
  const int E = in_sizes[1];                   // 3200000
  const long n4 = (long)M * F_OUT / 4;

  // 1) h = x @ w  (fp32 WMMA, 64x256 block tiles)
  gcn_gemm_wmma<<<(M + 63) / 64, 128, 0, stream>>>(x, w, h, M);
  // 2) out = 0
  gcn_zero<<<2048, 256, 0, stream>>>((v4f*)out, n4);
  // 3) out[r] += v * h[c]  (coalesced L2 atomics)
  gcn_spmm_atomic<<<(E + EPB - 1) / EPB, 256, 0, stream>>>(rows, cols, vals, h, out, E);
  // 4) ReLU in place
  gcn_relu<<<2048, 256, 0, stream>>>((v4f*)out, n4);
}